// SpaceAttention_68582037782608
// MI455X (gfx1250) — compile-verified
//
#include <hip/hip_runtime.h>
#include <math.h>

// ---------------------------------------------------------------------------
// SpaceAttention for MI455X (gfx1250, wave32).
//   S = 4*16*32*32 = 65536 tokens, E = 768, NH = 12, HEAD = 64.
//   Stage 1: qkv = x @ w_qkv + b_qkv, fused axial-RoPE epilogue  (WMMA f32)
//   Stage 2: per-token 12x12 attention over heads (LDS, VALU)
//   Stage 3: out = att @ w_proj + b_proj                         (WMMA f32)
// Workspace: qkv buffer 65536*2304 fp32 = 576 MiB (att written in-place over q).
// GEMM staging uses GLOBAL_LOAD_ASYNC_TO_LDS_B128 (ASYNCcnt) double-buffered.
// ---------------------------------------------------------------------------

typedef float v2f __attribute__((ext_vector_type(2)));
typedef float v8f __attribute__((ext_vector_type(8)));
typedef int   v4i __attribute__((ext_vector_type(4)));

#define K_DIM 768
#define A_STRIDE 20      // As[m][k] row stride (floats): 20m mod 64 distinct for m=0..15
#define B_STRIDE 136     // Bs[k][n] row stride (floats): 2*136 mod 64 == 16 -> halves disjoint
#define PI_F 3.14159265358979f
#define GT 4             // tokens per block in attention kernel

// ---- async global->LDS copy of 16 bytes per lane --------------------------
// Prototype (from toolchain diagnostic): (v4i addrspace(1)*, v4i addrspace(3)*, Imm, Imm)
#if __has_builtin(__builtin_amdgcn_global_load_async_to_lds_b128)
#define ASYNC_COPY16(dst_lds, src_glb)                                         \
    __builtin_amdgcn_global_load_async_to_lds_b128(                            \
        (__attribute__((address_space(1))) v4i*)(src_glb),                     \
        (__attribute__((address_space(3))) v4i*)(dst_lds), 0, 0)
#else
#define ASYNC_COPY16(dst_lds, src_glb)                                         \
    asm volatile("global_load_async_to_lds_b128 %0, %1, off"                   \
                 :: "v"((unsigned)(unsigned long long)                         \
                        (__attribute__((address_space(3))) void*)(dst_lds)),   \
                    "v"((unsigned long long)(src_glb))                         \
                 : "memory")
#endif

#if __has_builtin(__builtin_amdgcn_s_wait_asynccnt)
#define WAIT_ASYNC(n) __builtin_amdgcn_s_wait_asynccnt(n)
#else
#define WAIT_ASYNC(n) asm volatile("s_wait_asynccnt %0" :: "i"(n) : "memory")
#endif

// ---------------------------------------------------------------------------
// WMMA GEMM: C(MxN) = A(MxK) * B(KxN) + bias, K = 768 fixed.
// WG = 256 threads = 8 waves arranged 4(M) x 2(N); WG tile 128x128;
// per-wave tile 32x64 = 2x4 WMMA tiles (64 accumulator VGPRs / lane).
// ROPE==1: apply interleaved axial RoPE (token position from row index).
// ---------------------------------------------------------------------------
template <int ROPE>
__global__ __launch_bounds__(256)
void gemm_wmma_kernel(const float* __restrict__ A, int lda,
                      const float* __restrict__ B, int ldb,
                      const float* __restrict__ bias,
                      float* __restrict__ C, int ldc)
{
    __shared__ float As[2][128 * A_STRIDE];  // A tile row-major: As[m][k], padded
    __shared__ float Bs[2][16 * B_STRIDE];   // B tile: Bs[k][n], padded

    const int tid  = threadIdx.x;
    const int lane = tid & 31;
    const int wv   = tid >> 5;
    const int wvM  = wv & 3;       // 0..3
    const int wvN  = wv >> 2;      // 0..1
    const int hl   = lane >> 4;    // half-wave: 0 = lanes 0-15, 1 = lanes 16-31
    const int lr   = lane & 15;

    const int mBase = blockIdx.y * 128;
    const int nBase = blockIdx.x * 128;

    // issue one K-chunk (128x16 A tile + 16x128 B tile) of async copies:
    // 4 x b128 per thread -> 4 outstanding ASYNCcnt entries per wave.
    auto issue = [&](int kb, int buf) {
        #pragma unroll
        for (int p = 0; p < 2; ++p) {
            const int idx = tid + p * 256;       // 0..511
            const int m   = idx >> 2;            // 0..127
            const int kq  = (idx & 3) << 2;      // 0,4,8,12
            ASYNC_COPY16(&As[buf][m * A_STRIDE + kq],
                         &A[(size_t)(mBase + m) * lda + kb + kq]);
        }
        #pragma unroll
        for (int p = 0; p < 2; ++p) {
            const int idx = tid + p * 256;       // 0..511
            const int kr  = idx >> 5;            // 0..15
            const int nq  = (idx & 31) << 2;     // 0..124 step 4
            ASYNC_COPY16(&Bs[buf][kr * B_STRIDE + nq],
                         &B[(size_t)(kb + kr) * ldb + nBase + nq]);
        }
    };

    v8f acc[2][4] = {};

    const int NIT = K_DIM / 16;  // 48
    issue(0, 0);
    for (int it = 0; it < NIT; ++it) {
        const int buf = it & 1;
        if (it + 1 < NIT) {
            issue((it + 1) * 16, (it + 1) & 1);  // fill other buffer
            WAIT_ASYNC(4);                        // this chunk's 4 ops landed
        } else {
            WAIT_ASYNC(0);
        }
        __syncthreads();                          // all waves' fills visible

        #pragma unroll
        for (int kk = 0; kk < 16; kk += 4) {
            v2f a[2], b[4];
            #pragma unroll
            for (int i = 0; i < 2; ++i) {
                const int m = wvM * 32 + i * 16 + lr;
                // A operand (k, k+1) is contiguous in As[m][.] -> one ds_load_b64
                a[i] = *(const v2f*)&As[buf][m * A_STRIDE + kk + 2 * hl];
            }
            #pragma unroll
            for (int j = 0; j < 4; ++j) {
                const int n = wvN * 64 + j * 16 + lr;
                b[j].x = Bs[buf][(kk + 2 * hl + 0) * B_STRIDE + n];
                b[j].y = Bs[buf][(kk + 2 * hl + 1) * B_STRIDE + n];
            }
            #pragma unroll
            for (int i = 0; i < 2; ++i)
                #pragma unroll
                for (int j = 0; j < 4; ++j)
                    acc[i][j] = __builtin_amdgcn_wmma_f32_16x16x4_f32(
                        false, a[i], false, b[j], (short)0, acc[i][j],
                        false, false);
        }
        __syncthreads();   // all reads done before next fill overwrites buffer
    }

    // ---- epilogue: bias (+ RoPE), store ------------------------------------
    #pragma unroll
    for (int j = 0; j < 4; ++j) {
        const int nCol = nBase + wvN * 64 + j * 16 + lr;
        const float bv = bias[nCol];
        #pragma unroll
        for (int i = 0; i < 2; ++i) {
            #pragma unroll
            for (int r = 0; r < 8; ++r) {
                // C/D layout: VGPR r holds M = r (lo half) / r+8 (hi half), N = lane
                const int mRow = mBase + wvM * 32 + i * 16 + r + 8 * hl;
                float val = acc[i][j][r] + bv;
                if (ROPE) {
                    // partner channel (d^1) lives in the adjacent lane
                    const float partner = __shfl_xor(val, 1, 32);
                    const int d    = nCol & 63;          // channel within head
                    const int fidx = (d >> 1) & 15;      // base freq index
                    const int hh   = (mRow >> 5) & 31;   // token row in image
                    const int ww   = mRow & 31;          // token col in image
                    const int pos  = (d < 32) ? hh : ww; // axial: fh then fw
                    const float ax = -1.f + (2.f / 31.f) * (float)pos;
                    const float fr = PI_F * (1.f + (4.f / 15.f) * (float)fidx);
                    float sn, cs;
                    __sincosf(ax * fr, &sn, &cs);
                    val = (nCol & 1) ? (val * cs + partner * sn)
                                     : (val * cs - partner * sn);
                }
                C[(size_t)mRow * ldc + nCol] = val;
            }
        }
    }
}

// ---------------------------------------------------------------------------
// Per-token attention across the 12 heads (dim 64). Each block handles GT
// tokens: load 2304 floats/token to LDS, 12x12 scores, softmax, out = attn@v.
// Output is written in-place over the q region (cols 0..767) of the qkv row.
// ---------------------------------------------------------------------------
__global__ __launch_bounds__(256)
void attn_heads_kernel(float* __restrict__ qkv)
{
    __shared__ float tok[GT][2304];
    __shared__ float sc[GT][12][12];

    const int tid = threadIdx.x;
    const size_t s0 = (size_t)blockIdx.x * GT;

    // stage GT tokens' q/k/v rows
    for (int idx = tid; idx < GT * 576; idx += 256) {
        const int g  = idx / 576;
        const int q4 = (idx % 576) * 4;
        const float4 v = *(const float4*)(&qkv[(s0 + g) * 2304 + q4]);
        *(float4*)(&tok[g][q4]) = v;
    }
    __syncthreads();

    // scores[n][m] = q_n . k_m / sqrt(64)
    for (int t = tid; t < GT * 144; t += 256) {
        const int g = t / 144;
        const int r = t % 144;
        const int n = r / 12, m = r % 12;
        const float* qp = &tok[g][n * 64];
        const float* kp = &tok[g][768 + m * 64];
        float s = 0.f;
        #pragma unroll
        for (int i = 0; i < 64; ++i) s += qp[i] * kp[i];
        sc[g][n][m] = s * 0.125f;
    }
    __syncthreads();

    // softmax over m (rows of 12)
    for (int t = tid; t < GT * 12; t += 256) {
        const int g = t / 12, n = t % 12;
        float mx = -1e30f;
        #pragma unroll
        for (int m = 0; m < 12; ++m) mx = fmaxf(mx, sc[g][n][m]);
        float e[12], sum = 0.f;
        #pragma unroll
        for (int m = 0; m < 12; ++m) { e[m] = __expf(sc[g][n][m] - mx); sum += e[m]; }
        const float inv = 1.f / sum;
        #pragma unroll
        for (int m = 0; m < 12; ++m) sc[g][n][m] = e[m] * inv;
    }
    __syncthreads();

    // out[n][d] = sum_m attn[n][m] * v[m][d]  -> overwrite q region
    for (int t = tid; t < GT * 768; t += 256) {
        const int g = t / 768;
        const int c = t % 768;
        const int n = c >> 6;
        const int d = c & 63;
        float o = 0.f;
        #pragma unroll
        for (int m = 0; m < 12; ++m) o += sc[g][n][m] * tok[g][1536 + m * 64 + d];
        qkv[(s0 + g) * 2304 + c] = o;
    }
}

// ---------------------------------------------------------------------------
extern "C" void kernel_launch(void* const* d_in, const int* in_sizes, int n_in,
                              void* d_out, int out_size, void* d_ws, size_t ws_size,
                              hipStream_t stream)
{
    const float* x      = (const float*)d_in[0];  // (4,16,32,32,768)
    const float* w_qkv  = (const float*)d_in[1];  // (768, 2304)
    const float* b_qkv  = (const float*)d_in[2];  // (2304)
    const float* w_proj = (const float*)d_in[3];  // (768, 768)
    const float* b_proj = (const float*)d_in[4];  // (768)
    float* out = (float*)d_out;                   // (S, 768)
    float* qkv = (float*)d_ws;                    // needs 65536*2304*4 B

    const int S = 65536;
    dim3 blk(256);

    // Stage 1: qkv = x @ w_qkv + b, with RoPE epilogue. Grid 18 x 512.
    gemm_wmma_kernel<1><<<dim3(2304 / 128, S / 128), blk, 0, stream>>>(
        x, 768, w_qkv, 2304, b_qkv, qkv, 2304);

    // Stage 2: per-token attention over heads (writes att into q region).
    attn_heads_kernel<<<S / GT, blk, 0, stream>>>(qkv);

    // Stage 3: out = att @ w_proj + b. att rows live at stride 2304. Grid 6 x 512.
    gemm_wmma_kernel<0><<<dim3(768 / 128, S / 128), blk, 0, stream>>>(
        qkv, 2304, w_proj, 768, b_proj, out, 768);
}